// GAT_38603166056971
// MI455X (gfx1250) — compile-verified
//
#include <hip/hip_runtime.h>
#include <math.h>

// ---------------------------------------------------------------------------
// GAT 3-layer forward for MI455X (gfx1250, wave32).
//  - GEMMs use V_WMMA_F32_16X16X4_F32 (exact f32 precision, matches reference)
//    4 row tiles per wave; K and F are template constants so all loads are
//    base+immediate; software-pipelined so loads of step k+1 overlap the
//    4 WMMAs of step k.
//  - Edge phase uses hardware f32 atomics (L2-resident: node tables ~25.6MB
//    fit in the 192MB L2, so gathers/atomics should not touch HBM)
// ---------------------------------------------------------------------------

typedef float v2f __attribute__((ext_vector_type(2)));
typedef float v8f __attribute__((ext_vector_type(8)));

#define NEG_SLOPE 0.2f
#define EPS_F 1e-16f
#define H_HEADS 4

// ---------------- float ordered-uint mapping for atomic max ----------------
__device__ __forceinline__ unsigned fflip(float f) {
  unsigned u = __float_as_uint(f);
  return (u & 0x80000000u) ? ~u : (u | 0x80000000u);
}
__device__ __forceinline__ float funflip(unsigned u) {
  return (u & 0x80000000u) ? __uint_as_float(u ^ 0x80000000u)
                           : __uint_as_float(~u);
}

__device__ __forceinline__ void atomicAddF32(float* p, float v) {
  unsafeAtomicAdd(p, v);  // lowers to global_atomic_add_f32 on CDNA
}

// ---------------- zero fill (grid-stride) ----------------
__global__ void gat_zero_f32(float* __restrict__ p, long n) {
  long i = (long)blockIdx.x * blockDim.x + threadIdx.x;
  long stride = (long)gridDim.x * blockDim.x;
  for (; i < n; i += stride) p[i] = 0.0f;
}

// ---------------- WMMA f32 GEMM: out[N,F] = X[N,K] @ W[K,F] ----------------
// One wave computes a 64x16 output slab (4 stacked 16x16 tiles sharing one
// B fragment per K-step). K and F are compile-time so every load is
// base-pointer + immediate offset; the loop is fully unrolled and manually
// double-buffered (load step k+1, then issue the 4 WMMAs of step k).
template <int K, int F>
__global__ void gat_gemm_wmma_f32(const float* __restrict__ X,
                                  const float* __restrict__ W,
                                  float* __restrict__ out, int N) {
  const int colTiles = F >> 4;
  const int rowGroups = (N + 63) >> 6;  // 64 rows per wave
  int wave = blockIdx.x * (blockDim.x >> 5) + (threadIdx.x >> 5);
  if (wave >= rowGroups * colTiles) return;
  int rowGrp = wave / colTiles;
  int colTile = wave - rowGrp * colTiles;

  int lane = threadIdx.x & 31;
  int hf   = lane >> 4;   // 0: lanes 0-15, 1: lanes 16-31
  int l16  = lane & 15;

  int rowBase = rowGrp << 6;
  int ncol    = (colTile << 4) + l16;

  // A row base pointers (clamped for loads; stores stay guarded)
  int m0 = rowBase + l16;       if (m0 > N - 1) m0 = N - 1;
  int m1 = rowBase + 16 + l16;  if (m1 > N - 1) m1 = N - 1;
  int m2 = rowBase + 32 + l16;  if (m2 > N - 1) m2 = N - 1;
  int m3 = rowBase + 48 + l16;  if (m3 > N - 1) m3 = N - 1;
  const float* pA0 = X + (long)m0 * K + 2 * hf;
  const float* pA1 = X + (long)m1 * K + 2 * hf;
  const float* pA2 = X + (long)m2 * K + 2 * hf;
  const float* pA3 = X + (long)m3 * K + 2 * hf;
  // B base: b.x at pB[kk*F], b.y at pB[(kk+1)*F] (immediate offsets)
  const float* pB = W + (long)(2 * hf) * F + ncol;

  v8f c0 = {}, c1 = {}, c2 = {}, c3 = {};

  // prologue: load fragments of step 0
  v2f bc;
  bc.x = pB[0];
  bc.y = pB[F];
  float2 ac0 = *(const float2*)(pA0);
  float2 ac1 = *(const float2*)(pA1);
  float2 ac2 = *(const float2*)(pA2);
  float2 ac3 = *(const float2*)(pA3);

#pragma unroll
  for (int kk = 0; kk < K; kk += 4) {
    // load step kk+4 fragments (dead on the last iteration)
    v2f bn = {};
    float2 an0 = {}, an1 = {}, an2 = {}, an3 = {};
    if (kk + 4 < K) {
      bn.x = pB[(kk + 4) * F];
      bn.y = pB[(kk + 5) * F];
      an0 = *(const float2*)(pA0 + kk + 4);
      an1 = *(const float2*)(pA1 + kk + 4);
      an2 = *(const float2*)(pA2 + kk + 4);
      an3 = *(const float2*)(pA3 + kk + 4);
    }
    // issue the 4 WMMAs of step kk
    v2f a0; a0.x = ac0.x; a0.y = ac0.y;
    v2f a1; a1.x = ac1.x; a1.y = ac1.y;
    v2f a2; a2.x = ac2.x; a2.y = ac2.y;
    v2f a3; a3.x = ac3.x; a3.y = ac3.y;
    c0 = __builtin_amdgcn_wmma_f32_16x16x4_f32(false, a0, false, bc, (short)0,
                                               c0, false, false);
    c1 = __builtin_amdgcn_wmma_f32_16x16x4_f32(false, a1, false, bc, (short)0,
                                               c1, false, false);
    c2 = __builtin_amdgcn_wmma_f32_16x16x4_f32(false, a2, false, bc, (short)0,
                                               c2, false, false);
    c3 = __builtin_amdgcn_wmma_f32_16x16x4_f32(false, a3, false, bc, (short)0,
                                               c3, false, false);
    bc = bn;
    ac0 = an0; ac1 = an1; ac2 = an2; ac3 = an3;
  }

  // store: tile t rows [rowBase+16t+8*hf .. +8), column ncol
  int rb = rowBase + (hf << 3);
#pragma unroll
  for (int j = 0; j < 8; ++j) {
    int r0 = rb + j;
    if (r0 < N)      out[(long)r0 * F + ncol] = c0[j];
    int r1 = rb + 16 + j;
    if (r1 < N)      out[(long)r1 * F + ncol] = c1[j];
    int r2 = rb + 32 + j;
    if (r2 < N)      out[(long)r2 * F + ncol] = c2[j];
    int r3 = rb + 48 + j;
    if (r3 < N)      out[(long)r3 * F + ncol] = c3[j];
  }
}

// ------------- per-node attention coefficients + init of max/sum ----------
__global__ void gat_alpha_init(const float* __restrict__ hfeat,  // [N, H*C]
                               const float* __restrict__ a_src,  // [H*C]
                               const float* __restrict__ a_dst,  // [H*C]
                               float* __restrict__ alpha_s,      // [N, H]
                               float* __restrict__ alpha_d,      // [N, H]
                               unsigned* __restrict__ mflip,     // [N, H]
                               float* __restrict__ ssum,         // [N, H]
                               int N, int C) {
  int n = blockIdx.x * blockDim.x + threadIdx.x;
  if (n >= N) return;
  const float4* row = (const float4*)(hfeat + (long)n * (H_HEADS * C));
#pragma unroll
  for (int h = 0; h < H_HEADS; ++h) {
    const float4* rs = row + h * (C / 4);
    const float4* vs = (const float4*)(a_src + h * C);
    const float4* vd = (const float4*)(a_dst + h * C);
    float as = 0.0f, ad = 0.0f;
    for (int c4 = 0; c4 < C / 4; ++c4) {
      float4 v = rs[c4];
      float4 s = vs[c4];
      float4 d = vd[c4];
      as = fmaf(v.x, s.x, fmaf(v.y, s.y, fmaf(v.z, s.z, fmaf(v.w, s.w, as))));
      ad = fmaf(v.x, d.x, fmaf(v.y, d.y, fmaf(v.z, d.z, fmaf(v.w, d.w, ad))));
    }
    alpha_s[n * H_HEADS + h] = as;
    alpha_d[n * H_HEADS + h] = ad;
    mflip[n * H_HEADS + h]  = 0u;     // below fflip of any finite float
    ssum[n * H_HEADS + h]   = 0.0f;
  }
}

// ---------------- edge pass 1: segment max of leaky-relu logits ------------
__global__ void gat_edge_max(const long long* __restrict__ srcA,
                             const long long* __restrict__ dstA,
                             const float* __restrict__ alpha_s,
                             const float* __restrict__ alpha_d,
                             unsigned* __restrict__ mflip,
                             int E, int N) {
  int i = blockIdx.x * blockDim.x + threadIdx.x;
  if (i >= E + N) return;
  int s = (i < E) ? (int)srcA[i] : (i - E);
  int d = (i < E) ? (int)dstA[i] : (i - E);
  float4 as4 = *(const float4*)(alpha_s + (long)s * H_HEADS);
  float4 ad4 = *(const float4*)(alpha_d + (long)d * H_HEADS);
  float e[4] = {as4.x + ad4.x, as4.y + ad4.y, as4.z + ad4.z, as4.w + ad4.w};
#pragma unroll
  for (int h = 0; h < H_HEADS; ++h) {
    float v = (e[h] > 0.0f) ? e[h] : NEG_SLOPE * e[h];
    atomicMax(&mflip[d * H_HEADS + h], fflip(v));
  }
}

// ------- edge pass 2: exp(e - max) per edge, segment sum via atomics -------
__global__ void gat_edge_exp(const long long* __restrict__ srcA,
                             const long long* __restrict__ dstA,
                             const float* __restrict__ alpha_s,
                             const float* __restrict__ alpha_d,
                             const unsigned* __restrict__ mflip,
                             float* __restrict__ ebuf,   // [E+N, H]
                             float* __restrict__ ssum,   // [N, H]
                             int E, int N) {
  int i = blockIdx.x * blockDim.x + threadIdx.x;
  if (i >= E + N) return;
  int s = (i < E) ? (int)srcA[i] : (i - E);
  int d = (i < E) ? (int)dstA[i] : (i - E);
  float4 as4 = *(const float4*)(alpha_s + (long)s * H_HEADS);
  float4 ad4 = *(const float4*)(alpha_d + (long)d * H_HEADS);
  float e[4] = {as4.x + ad4.x, as4.y + ad4.y, as4.z + ad4.z, as4.w + ad4.w};
  float4 ex4;
#pragma unroll
  for (int h = 0; h < H_HEADS; ++h) {
    float v = (e[h] > 0.0f) ? e[h] : NEG_SLOPE * e[h];
    float ex = expf(v - funflip(mflip[d * H_HEADS + h]));
    (&ex4.x)[h] = ex;
    atomicAddF32(&ssum[d * H_HEADS + h], ex);
  }
  *(float4*)(ebuf + (long)i * H_HEADS) = ex4;
}

// ------- edge pass 3: one wave per edge, scatter-add weighted message ------
__global__ void gat_edge_aggr(const long long* __restrict__ srcA,
                              const long long* __restrict__ dstA,
                              const float* __restrict__ hfeat,  // [N, F]
                              const float* __restrict__ ebuf,   // [E+N, H]
                              const float* __restrict__ ssum,   // [N, H]
                              float* __restrict__ acc,          // [N, F]
                              int E, int N, int F, int C) {
  int i = blockIdx.x * (blockDim.x >> 5) + (threadIdx.x >> 5);
  if (i >= E + N) return;
  int lane = threadIdx.x & 31;
  int s = (i < E) ? (int)srcA[i] : (i - E);
  int d = (i < E) ? (int)dstA[i] : (i - E);
  // per-head attention weights: two b128 loads, broadcast across the wave
  float4 e4 = *(const float4*)(ebuf + (long)i * H_HEADS);
  float4 s4 = *(const float4*)(ssum + (long)d * H_HEADS);
  float att[4];
  att[0] = e4.x / (s4.x + EPS_F);
  att[1] = e4.y / (s4.y + EPS_F);
  att[2] = e4.z / (s4.z + EPS_F);
  att[3] = e4.w / (s4.w + EPS_F);

  const float* hrow = hfeat + (long)s * F;
  float* arow = acc + (long)d * F;
  for (int c = lane; c < F; c += 32) {
    int h = c / C;  // C is a multiple of 32, lane<32: head constant per step
    atomicAddF32(&arow[c], hrow[c] * att[h]);
  }
}

// ---------------- finalize: +bias, relu (layers 0 and 1) ------------------
__global__ void gat_finalize_relu(const float* __restrict__ acc,
                                  const float* __restrict__ bias,
                                  float* __restrict__ outp,
                                  long total, int F) {
  long i = (long)blockIdx.x * blockDim.x + threadIdx.x;
  if (i >= total) return;
  int j = (int)(i % F);
  float v = acc[i] + bias[j];
  outp[i] = v > 0.0f ? v : 0.0f;
}

// ---------------- finalize: head mean + bias (layer 2) --------------------
__global__ void gat_finalize_mean(const float* __restrict__ acc,  // [N, H*Co]
                                  const float* __restrict__ b2,   // [Co]
                                  float* __restrict__ outp,       // [N, Co]
                                  int N, int Co) {
  int i = blockIdx.x * blockDim.x + threadIdx.x;
  if (i >= N * Co) return;
  int n = i / Co, c = i - n * Co;
  const float* r = acc + (long)n * (H_HEADS * Co);
  float v = 0.25f * (r[c] + r[Co + c] + r[2 * Co + c] + r[3 * Co + c]) + b2[c];
  outp[i] = v;
}

// ---------------------------------------------------------------------------
// Host side
// ---------------------------------------------------------------------------
static inline long alignUp(long x) { return (x + 255) & ~255L; }

struct GatScratch {
  float* lin;       // [N, 256]  gemm output (current layer)
  float* acc;       // [N, 256]  aggregation accumulator
  float* h1;        // [N, 128]  layer-0 output
  float* h2;        // [N, 128]  layer-1 output
  float* alpha_s;   // [N, 4]
  float* alpha_d;   // [N, 4]
  unsigned* mflip;  // [N, 4]
  float* ssum;      // [N, 4]
  float* ebuf;      // [E+N, 4]
};

static void run_layer(const float* Xin, const float* W, const float* a_s,
                      const float* a_d, const long long* src,
                      const long long* dst, const GatScratch& S,
                      int N, int E, int Fout, int C, hipStream_t stream) {
  // 1) lin = Xin @ W  (WMMA; K fixed at 128, F is 128 or 256 in this network)
  {
    int waves = ((N + 63) >> 6) * (Fout >> 4);
    int blocks = (waves + 7) / 8;  // 8 waves (256 threads) per block
    if (Fout == 128)
      gat_gemm_wmma_f32<128, 128><<<blocks, 256, 0, stream>>>(Xin, W, S.lin, N);
    else
      gat_gemm_wmma_f32<128, 256><<<blocks, 256, 0, stream>>>(Xin, W, S.lin, N);
  }
  // 2) per-node alphas + init m/s
  gat_alpha_init<<<(N + 255) / 256, 256, 0, stream>>>(
      S.lin, a_s, a_d, S.alpha_s, S.alpha_d, S.mflip, S.ssum, N, C);
  // 3) zero accumulator
  gat_zero_f32<<<2048, 256, 0, stream>>>(S.acc, (long)N * Fout);
  // 4) edge passes
  int EN = E + N;
  gat_edge_max<<<(EN + 255) / 256, 256, 0, stream>>>(
      src, dst, S.alpha_s, S.alpha_d, S.mflip, E, N);
  gat_edge_exp<<<(EN + 255) / 256, 256, 0, stream>>>(
      src, dst, S.alpha_s, S.alpha_d, S.mflip, S.ebuf, S.ssum, E, N);
  gat_edge_aggr<<<(EN + 7) / 8, 256, 0, stream>>>(
      src, dst, S.lin, S.ebuf, S.ssum, S.acc, E, N, Fout, C);
}

extern "C" void kernel_launch(void* const* d_in, const int* in_sizes, int n_in,
                              void* d_out, int out_size, void* d_ws,
                              size_t ws_size, hipStream_t stream) {
  const float*     x    = (const float*)d_in[0];
  const long long* eidx = (const long long*)d_in[1];  // int64 [2, E]
  const float* W0 = (const float*)d_in[2];
  const float* as0 = (const float*)d_in[3];
  const float* ad0 = (const float*)d_in[4];
  const float* b0 = (const float*)d_in[5];
  const float* W1 = (const float*)d_in[6];
  const float* as1 = (const float*)d_in[7];
  const float* ad1 = (const float*)d_in[8];
  const float* b1 = (const float*)d_in[9];
  const float* W2 = (const float*)d_in[10];
  const float* as2 = (const float*)d_in[11];
  const float* ad2 = (const float*)d_in[12];
  const float* b2 = (const float*)d_in[13];
  float* out = (float*)d_out;

  const int F_IN = 128, F_HID = 128, F_L2 = 256, C_HID = 32, C_OUT = 64;
  int N = in_sizes[0] / F_IN;
  int E = in_sizes[1] / 2;
  const long long* src = eidx;
  const long long* dst = eidx + E;

  // workspace carve-up
  char* p = (char*)d_ws;
  GatScratch S;
  S.lin = (float*)p;       p += alignUp((long)N * 256 * 4);
  S.acc = (float*)p;       p += alignUp((long)N * 256 * 4);
  S.h1  = (float*)p;       p += alignUp((long)N * 128 * 4);
  S.h2  = (float*)p;       p += alignUp((long)N * 128 * 4);
  S.alpha_s = (float*)p;   p += alignUp((long)N * 4 * 4);
  S.alpha_d = (float*)p;   p += alignUp((long)N * 4 * 4);
  S.mflip = (unsigned*)p;  p += alignUp((long)N * 4 * 4);
  S.ssum = (float*)p;      p += alignUp((long)N * 4 * 4);
  S.ebuf = (float*)p;      p += alignUp((long)(E + N) * 4 * 4);

  // ---- layer 0: x -> h1 (concat, relu) ----
  run_layer(x, W0, as0, ad0, src, dst, S, N, E, F_HID, C_HID, stream);
  gat_finalize_relu<<<(int)(((long)N * F_HID + 255) / 256), 256, 0, stream>>>(
      S.acc, b0, S.h1, (long)N * F_HID, F_HID);

  // ---- layer 1: h1 -> h2 (concat, relu) ----
  run_layer(S.h1, W1, as1, ad1, src, dst, S, N, E, F_HID, C_HID, stream);
  gat_finalize_relu<<<(int)(((long)N * F_HID + 255) / 256), 256, 0, stream>>>(
      S.acc, b1, S.h2, (long)N * F_HID, F_HID);

  // ---- layer 2: h2 -> out (head mean + bias) ----
  run_layer(S.h2, W2, as2, ad2, src, dst, S, N, E, F_L2, C_OUT, stream);
  gat_finalize_mean<<<(N * C_OUT + 255) / 256, 256, 0, stream>>>(
      S.acc, b2, out, N, C_OUT);
}